// SolveLS_14714557956195
// MI455X (gfx1250) — compile-verified
//
#include <hip/hip_runtime.h>
#include <math.h>

typedef __attribute__((ext_vector_type(2))) float v2f;
typedef __attribute__((ext_vector_type(8))) float v8f;

#define NB    16
#define CIN   32
#define COUT  64
#define HH    128
#define WWID  128
#define DS    5
#define KDIM  800            // CIN*DS*DS
#define HW    16384          // HH*WWID
#define QSTRIDE (KDIM*KDIM)
#define PW    136            // padded width  (pad 4 each side)
#define PADHW (PW*PW)        // 18496 floats per padded channel

// -------------------------------------------------------------------------
// Kernel 0: zero-padded copy of x:  xpad[n][c][136][136]
// -------------------------------------------------------------------------
__global__ void pad_kernel(const float* __restrict__ x, float* __restrict__ xpad) {
  const int total = NB * CIN * PADHW;
  for (int idx = blockIdx.x * blockDim.x + threadIdx.x; idx < total;
       idx += gridDim.x * blockDim.x) {
    int wp = idx % PW;
    int t  = idx / PW;
    int hp = t % PW;
    int c2 = t / PW;                 // n*CIN + c
    int h = hp - 4, w = wp - 4;
    float v = ((unsigned)h < (unsigned)HH && (unsigned)w < (unsigned)WWID)
                  ? x[(long)c2 * HW + h * WWID + w] : 0.0f;
    xpad[idx] = v;
  }
}

// -------------------------------------------------------------------------
// Kernel 1: R[n,i,j,u,v] = sum_{h,w} x[n,i,h,w] * x[n,j,h+u-4,w+v-4]
// One block per (n,u,v); 4 waves, each a 16x16 tile of the 32x32 channel
// correlation. f32 WMMA 16x16x4, K = 16384, unconditional b64 loads from
// the padded image. Two accumulators break the WMMA RAW chain on C.
// Fragment map (ISA 7.12.2): lane L -> row/col = L%16, K = (L>=16?2:0)+vgpr
// -------------------------------------------------------------------------
__global__ __launch_bounds__(128) void r_kernel(const float* __restrict__ xpad,
                                                float* __restrict__ R) {
  int b  = blockIdx.x;           // n*81 + u*9 + v
  int n  = b / 81;
  int uv = b % 81;
  int su = uv / 9 - 4;
  int sv = uv % 9 - 4;

  int lane  = threadIdx.x & 31;  // wave32
  int wave  = threadIdx.x >> 5;  // 0..3
  int ti    = (wave >> 1) * 16;
  int tj    = (wave & 1) * 16;
  int mrow  = lane & 15;
  int khalf = (lane >> 4) * 2;

  const float* xa = xpad + ((long)n * CIN + (ti + mrow)) * PADHW;
  const float* xb = xpad + ((long)n * CIN + (tj + mrow)) * PADHW;

  v8f acc0 = {}, acc1 = {};
  for (int h = 0; h < HH; ++h) {
    const float* pa = xa + (h + 4) * PW + 4 + khalf;
    const float* pb = xb + (h + 4 + su) * PW + (4 + sv) + khalf;
#pragma unroll 4
    for (int w0 = 0; w0 < WWID; w0 += 8) {
      v2f a0 = *(const v2f*)(pa + w0);
      v2f b0 = *(const v2f*)(pb + w0);
      acc0 = __builtin_amdgcn_wmma_f32_16x16x4_f32(false, a0, false, b0,
                                                   (short)0, acc0, false, false);
      v2f a1 = *(const v2f*)(pa + w0 + 4);
      v2f b1 = *(const v2f*)(pb + w0 + 4);
      acc1 = __builtin_amdgcn_wmma_f32_16x16x4_f32(false, a1, false, b1,
                                                   (short)0, acc1, false, false);
    }
  }

  // D layout: vgpr r, lane L -> row = r + (L>=16?8:0), col = L%16
  int col    = tj + mrow;
  int rowadd = (lane >> 4) * 8;
#pragma unroll
  for (int r = 0; r < 8; ++r) {
    int row = ti + r + rowadd;
    R[(((long)n * CIN + row) * CIN + col) * 81 + uv] = acc0[r] + acc1[r];
  }
}

// -------------------------------------------------------------------------
// Kernel 2: gather R -> Q (block-Toeplitz expansion) + a*I on the diagonal
// -------------------------------------------------------------------------
__global__ void q_kernel(const float* __restrict__ R,
                         const float* __restrict__ alpha,
                         const float* __restrict__ reg,
                         float* __restrict__ Q) {
  const long total = (long)NB * QSTRIDE;
  float rg = reg[0];
  for (long idx = (long)blockIdx.x * blockDim.x + threadIdx.x; idx < total;
       idx += (long)gridDim.x * blockDim.x) {
    int n   = (int)(idx / QSTRIDE);
    int rem = (int)(idx % QSTRIDE);
    int row = rem / KDIM, col = rem % KDIM;
    int i = row / 25, r1 = row % 25, di = r1 / 5, dj = r1 % 5;
    int j = col / 25, r2 = col % 25, di2 = r2 / 5, dj2 = r2 % 5;
    int u = di2 - di + 4, v = dj2 - dj + 4;
    float q = R[(((long)n * CIN + i) * CIN + j) * 81 + u * 9 + v];
    if (row == col) q += alpha[n] * ((float)HW * rg / (float)KDIM);
    Q[idx] = q;
  }
}

// -------------------------------------------------------------------------
// Kernel 3: P[n,(j,di,dj),o] = sum_{h,w} x[n,j,h+di-2,w+dj-2]*y[n,o,h,w]
//            + a * d[n,o,(j,di,dj)]   (f32 WMMA, 50x4 tiles per batch,
//            A from padded image -> unconditional loads)
// -------------------------------------------------------------------------
__global__ __launch_bounds__(128) void p_kernel(const float* __restrict__ xpad,
                                                const float* __restrict__ y,
                                                const float* __restrict__ dmat,
                                                const float* __restrict__ alpha,
                                                const float* __restrict__ reg,
                                                float* __restrict__ P) {
  int n  = blockIdx.x / 50;
  int mt = blockIdx.x % 50;
  int lane = threadIdx.x & 31;
  int nt   = threadIdx.x >> 5;          // 4 waves -> 4 col tiles (COUT=64)

  int m  = mt * 16 + (lane & 15);       // P row = (j, di, dj)
  int j  = m / 25;
  int rm = m % 25;
  int di = rm / 5 - 2;
  int dj = rm % 5 - 2;
  const float* xr = xpad + ((long)n * CIN + j) * PADHW;

  int o = nt * 16 + (lane & 15);
  const float* yr = y + ((long)n * COUT + o) * HW;
  int khalf = (lane >> 4) * 2;

  v8f acc0 = {}, acc1 = {};
  for (int h = 0; h < HH; ++h) {
    const float* pa = xr + (h + 4 + di) * PW + (4 + dj) + khalf;
    const float* pb = yr + h * WWID + khalf;
#pragma unroll 4
    for (int w0 = 0; w0 < WWID; w0 += 8) {
      v2f a0 = *(const v2f*)(pa + w0);
      v2f b0 = *(const v2f*)(pb + w0);
      acc0 = __builtin_amdgcn_wmma_f32_16x16x4_f32(false, a0, false, b0,
                                                   (short)0, acc0, false, false);
      v2f a1 = *(const v2f*)(pa + w0 + 4);
      v2f b1 = *(const v2f*)(pb + w0 + 4);
      acc1 = __builtin_amdgcn_wmma_f32_16x16x4_f32(false, a1, false, b1,
                                                   (short)0, acc1, false, false);
    }
  }

  float an = alpha[n] * ((float)HW * reg[0] / (float)KDIM);
  int oc     = nt * 16 + (lane & 15);
  int rowadd = (lane >> 4) * 8;
#pragma unroll
  for (int r = 0; r < 8; ++r) {
    int row = mt * 16 + r + rowadd;
    float v = acc0[r] + acc1[r] + an * dmat[((long)n * COUT + oc) * KDIM + row];
    P[((long)n * KDIM + row) * COUT + oc] = v;
  }
}

// -------------------------------------------------------------------------
// Kernel 4: in-place lower Cholesky, one workgroup per batch.
// -------------------------------------------------------------------------
__global__ __launch_bounds__(256) void chol_kernel(float* __restrict__ Q) {
  float* A = Q + (long)blockIdx.x * QSTRIDE;
  __shared__ float col[KDIM];
  __shared__ float sdiag;
  int tid = threadIdx.x, nt = blockDim.x;
  for (int k = 0; k < KDIM; ++k) {
    if (tid == 0) {
      float dv = sqrtf(A[(long)k * KDIM + k]);
      A[(long)k * KDIM + k] = dv;
      sdiag = dv;
    }
    __syncthreads();
    float dinv = 1.0f / sdiag;
    for (int i = k + 1 + tid; i < KDIM; i += nt) {
      float v = A[(long)i * KDIM + k] * dinv;
      A[(long)i * KDIM + k] = v;
      col[i] = v;
    }
    __syncthreads();
    for (int i = k + 1 + tid; i < KDIM; i += nt) {
      float lik = col[i];
      float* rowp = A + (long)i * KDIM;
      for (int jj = k + 1; jj <= i; ++jj) rowp[jj] -= lik * col[jj];
    }
    __syncthreads();
  }
}

// -------------------------------------------------------------------------
// Kernel 5: forward (L z = P) then backward (L^T D = z), in place over P.
// -------------------------------------------------------------------------
__global__ __launch_bounds__(64) void solve_kernel(const float* __restrict__ Q,
                                                   float* __restrict__ P) {
  int n = blockIdx.x;
  const float* L = Q + (long)n * QSTRIDE;
  float* Z = P + (long)n * KDIM * COUT;
  int c = threadIdx.x;
  __shared__ float lrow[KDIM];

  for (int i = 0; i < KDIM; ++i) {
    for (int jj = c; jj < i; jj += 64) lrow[jj] = L[(long)i * KDIM + jj];
    __syncthreads();
    float s = Z[(long)i * COUT + c];
    for (int jj = 0; jj < i; ++jj) s -= lrow[jj] * Z[(long)jj * COUT + c];
    Z[(long)i * COUT + c] = s / L[(long)i * KDIM + i];
    __syncthreads();
  }
  for (int i = KDIM - 1; i >= 0; --i) {
    for (int jj = i + 1 + c; jj < KDIM; jj += 64) lrow[jj] = L[(long)jj * KDIM + i];
    __syncthreads();
    float s = Z[(long)i * COUT + c];
    for (int jj = i + 1; jj < KDIM; ++jj) s -= lrow[jj] * Z[(long)jj * COUT + c];
    Z[(long)i * COUT + c] = s / L[(long)i * KDIM + i];
    __syncthreads();
  }
}

// -------------------------------------------------------------------------
// Kernel 6: out[n,o,i,di,dj] = Dsol[n,(i,di,dj),o]
// -------------------------------------------------------------------------
__global__ void out_kernel(const float* __restrict__ P, float* __restrict__ out) {
  const int total = NB * COUT * KDIM;
  for (int idx = blockIdx.x * blockDim.x + threadIdx.x; idx < total;
       idx += gridDim.x * blockDim.x) {
    int n   = idx / (COUT * KDIM);
    int rem = idx % (COUT * KDIM);
    int o = rem / KDIM, row = rem % KDIM;
    out[idx] = P[((long)n * KDIM + row) * COUT + o];
  }
}

extern "C" void kernel_launch(void* const* d_in, const int* in_sizes, int n_in,
                              void* d_out, int out_size, void* d_ws, size_t ws_size,
                              hipStream_t stream) {
  const float* x     = (const float*)d_in[0];   // (16,1,32,128,128)
  const float* dmat  = (const float*)d_in[1];   // (16,64,32,5,5)
  const float* y     = (const float*)d_in[2];   // (16,64,1,128,128)
  const float* alpha = (const float*)d_in[3];   // (16,1,1,1)
  const float* reg   = (const float*)d_in[4];   // scalar
  float* out = (float*)d_out;

  // Workspace layout (floats). xpad aliases the Q region: xpad is consumed by
  // r_kernel/p_kernel and dead before q_kernel writes Q (same stream, ordered).
  float* Q    = (float*)d_ws;                   // 16*800*800   = 10,240,000 f
  float* xpad = Q;                              //  16*32*136*136 = 9,469,952 f (<= Q)
  float* R    = Q + (long)NB * QSTRIDE;         // 16*32*32*81  =  1,327,104 f
  float* P    = R + (long)NB * CIN * CIN * 81;  // 16*800*64    =    819,200 f

  pad_kernel <<<dim3(4096),    dim3(256), 0, stream>>>(x, xpad);
  r_kernel   <<<dim3(NB * 81), dim3(128), 0, stream>>>(xpad, R);
  p_kernel   <<<dim3(NB * 50), dim3(128), 0, stream>>>(xpad, y, dmat, alpha, reg, P);
  q_kernel   <<<dim3(4096),    dim3(256), 0, stream>>>(R, alpha, reg, Q);
  chol_kernel<<<dim3(NB),      dim3(256), 0, stream>>>(Q);
  solve_kernel<<<dim3(NB),     dim3(64),  0, stream>>>(Q, P);
  out_kernel <<<dim3(2048),    dim3(256), 0, stream>>>(P, out);
}